// SSKernelNPLR_31310311588115
// MI455X (gfx1250) — compile-verified
//
#include <hip/hip_runtime.h>
#include <math.h>

typedef __attribute__((ext_vector_type(2))) float v2f;
typedef __attribute__((ext_vector_type(8))) float v8f;

#define HH 512
#define NN 32
#define LL 4096
#define MCOUNT 256     // LL / 16  (one "m" row per thread)
#define KDIM   64      // 2*NN: [0,32) = Re part, [32,64) = -Im part
#define A_STRIDE 68    // padded row stride (floats) to dodge LDS bank conflicts

// K[h, 16*m + j] = sum_{n'} A[m,n'] * B[n',j]
//   A[m,n]    =  Re(G_n * P_{n,m}),  A[m,n+32] = -Im(G_n * P_{n,m})
//   B[n,j]    =  Re(Q_{n,j}),        B[n+32,j] =  Im(Q_{n,j})
//   P = exp(16*z*m), Q = exp(z*j), G = 2*C*Bbar
__global__ __launch_bounds__(256)
void ssm_vandermonde_wmma(const float* __restrict__ log_dt,
                          const float* __restrict__ Lam_re,
                          const float* __restrict__ Lam_im,
                          const float* __restrict__ B_re,
                          const float* __restrict__ B_im,
                          const float* __restrict__ C_re,
                          const float* __restrict__ C_im,
                          const float* __restrict__ Dskip,
                          float* __restrict__ Kout)
{
    __shared__ float sZre[NN], sZim[NN], sGre[NN], sGim[NN];
    __shared__ float sB[KDIM][16];            // 4 KB
    __shared__ float sA[MCOUNT][A_STRIDE];    // ~68 KB

    const int h   = blockIdx.x;
    const int tid = threadIdx.x;

    // ---- Step 1: per-state discretization (lanes 0..31, one n each) ----
    if (tid < NN) {
        const int n = tid;
        const float dt  = __expf(log_dt[h]);
        const float lre = Lam_re[h * NN + n];
        const float lim = Lam_im[h * NN + n];
        const float zre = lre * dt, zim = lim * dt;
        // Lambda_bar = exp(z)
        float er = __expf(zre);
        float sn, cs; __sincosf(zim, &sn, &cs);
        const float lbre = er * cs, lbim = er * sn;
        // (Lambda_bar - 1) / Lambda   (complex divide via conj/|.|^2)
        const float nre = lbre - 1.0f, nim = lbim;
        const float inv = 1.0f / (lre * lre + lim * lim);
        const float qre = (nre * lre + nim * lim) * inv;
        const float qim = (nim * lre - nre * lim) * inv;
        // Bbar = q * B
        const float bre = B_re[h * NN + n], bim = B_im[h * NN + n];
        const float bbre = qre * bre - qim * bim;
        const float bbim = qre * bim + qim * bre;
        // G = 2 * C * Bbar
        const float cre = C_re[h * NN + n], cim = C_im[h * NN + n];
        sGre[n] = 2.0f * (cre * bbre - cim * bbim);
        sGim[n] = 2.0f * (cre * bbim + cim * bbre);
        sZre[n] = zre;
        sZim[n] = zim;
    }
    __syncthreads();

    // ---- Step 2: B matrix, Q[n,j] = exp(z_n * j), j in [0,16) ----
    for (int idx = tid; idx < NN * 16; idx += 256) {
        const int n = idx & (NN - 1);
        const int j = idx >> 5;
        const float fj = (float)j;
        const float er = __expf(sZre[n] * fj);
        float sn, cs; __sincosf(sZim[n] * fj, &sn, &cs);
        sB[n][j]      = er * cs;   // Re(Q)
        sB[n + NN][j] = er * sn;   // Im(Q)
    }

    // ---- Step 3: A matrix, thread tid owns row m = tid ----
    {
        const int m = tid;
        const float fm = (float)(16 * m);
        #pragma unroll 4
        for (int n = 0; n < NN; ++n) {
            // args up to ~1600 rad: use range-reducing sincosf
            const float er = expf(sZre[n] * fm);
            float sn, cs; sincosf(sZim[n] * fm, &sn, &cs);
            const float pre = er * cs, pim = er * sn;
            const float gre = sGre[n], gim = sGim[n];
            sA[m][n]      =  gre * pre - gim * pim;   //  Re(G*P)
            sA[m][n + NN] = -(gre * pim + gim * pre); // -Im(G*P)
        }
    }
    __syncthreads();

    // ---- Step 4: GEMM via V_WMMA_F32_16X16X4_F32 ----
    // wave w handles m-tiles 2w and 2w+1; each tile: 16 k-steps of K=4
    const int wave = tid >> 5;
    const int lane = tid & 31;
    const int half = lane >> 4;   // 0: lanes 0-15, 1: lanes 16-31
    const int lrow = lane & 15;   // A row (M) for A-frag; column (N=j) for B/D

    float* outh = Kout + (size_t)h * LL;
    const float dval = Dskip[h];

    #pragma unroll
    for (int t = 0; t < 2; ++t) {
        const int mt    = wave * 2 + t;
        const int Mbase = mt * 16;
        v8f acc = (v8f){0.f, 0.f, 0.f, 0.f, 0.f, 0.f, 0.f, 0.f};
        #pragma unroll
        for (int ks = 0; ks < 16; ++ks) {
            const int kb = ks * 4 + half * 2;
            // A 16x4 layout: lane L holds row M=L%16 (M = lane&15);
            //   VGPR0 = K = 2*half, VGPR1 = K = 2*half+1
            v2f a;
            a.x = sA[Mbase + lrow][kb];
            a.y = sA[Mbase + lrow][kb + 1];
            // B 4x16 layout (mirror): lane holds column N = lane&15;
            //   VGPR0 = row K = 2*half, VGPR1 = row K = 2*half+1
            v2f b;
            b.x = sB[kb][lrow];
            b.y = sB[kb + 1][lrow];
            acc = __builtin_amdgcn_wmma_f32_16x16x4_f32(
                false, a, false, b, (short)0, acc, false, false);
        }
        // D 16x16 layout: lane holds column N = lane&15; VGPR v = row (v + 8*half)
        #pragma unroll
        for (int v = 0; v < 8; ++v) {
            const int m = Mbase + v + 8 * half;
            const int l = 16 * m + lrow;
            float val = acc[v];
            if (l == 0) val += dval;   // D[h] only at lag 0
            outh[l] = val;
        }
    }
}

extern "C" void kernel_launch(void* const* d_in, const int* in_sizes, int n_in,
                              void* d_out, int out_size, void* d_ws, size_t ws_size,
                              hipStream_t stream) {
    (void)in_sizes; (void)n_in; (void)d_ws; (void)ws_size; (void)out_size;
    const float* log_dt = (const float*)d_in[0];
    const float* lam_re = (const float*)d_in[1];
    const float* lam_im = (const float*)d_in[2];
    const float* b_re   = (const float*)d_in[3];
    const float* b_im   = (const float*)d_in[4];
    const float* c_re   = (const float*)d_in[5];
    const float* c_im   = (const float*)d_in[6];
    const float* dvec   = (const float*)d_in[7];
    float* out = (float*)d_out;

    ssm_vandermonde_wmma<<<dim3(HH), dim3(256), 0, stream>>>(
        log_dt, lam_re, lam_im, b_re, b_im, c_re, c_im, dvec, out);
}